// GNN_5033701671377
// MI455X (gfx1250) — compile-verified
//
#include <hip/hip_runtime.h>

#define NN 100000
#define NE 1600000
#define NG 64
#define INF_ 10
#define HIDF 128
#define OUTF 101

typedef __attribute__((ext_vector_type(2))) float v2f;
typedef __attribute__((ext_vector_type(8))) float v8f;

// ---------------- utility fills ----------------
__global__ void fill_kernel(float* __restrict__ p, float v, int n) {
    int i = blockIdx.x * blockDim.x + threadIdx.x;
    if (i < n) p[i] = v;
}

// ---------------- gcn_norm ----------------
__global__ void degree_kernel(const int* __restrict__ dst, float* __restrict__ deg, int E) {
    int e = blockIdx.x * blockDim.x + threadIdx.x;
    if (e < E) unsafeAtomicAdd(&deg[dst[e]], 1.0f);
}

__global__ void norm_kernel(const float* __restrict__ deg, float* __restrict__ dis,
                            float* __restrict__ invdeg, int n) {
    int i = blockIdx.x * blockDim.x + threadIdx.x;
    if (i < n) {
        float d = deg[i];              // >= 1 (self-loop included)
        dis[i]    = rsqrtf(d);
        invdeg[i] = 1.0f / d;          // self-loop edge weight dis[i]^2
    }
}

__global__ void edgew_kernel(const int* __restrict__ src, const int* __restrict__ dst,
                             const float* __restrict__ dis, float* __restrict__ w, int E) {
    int e = blockIdx.x * blockDim.x + threadIdx.x;
    if (e < E) w[e] = dis[src[e]] * dis[dst[e]];
}

// ---------------- layer-1 transform (K=10, scalar; trivial FLOPs) ----------------
__global__ __launch_bounds__(256) void gemm_in_kernel(const float* __restrict__ X,
                                                      const float* __restrict__ W1,
                                                      float* __restrict__ H, int M) {
    __shared__ float Ws[INF_ * HIDF];
    for (int i = threadIdx.x; i < INF_ * HIDF; i += 256) Ws[i] = W1[i];
    __syncthreads();
    int gid = blockIdx.x * blockDim.x + threadIdx.x;
    if (gid >= M * HIDF) return;
    int i = gid >> 7, f = gid & 127;
    const float* xr = X + (size_t)i * INF_;
    float acc = 0.0f;
#pragma unroll
    for (int k = 0; k < INF_; ++k) acc = fmaf(xr[k], Ws[k * HIDF + f], acc);
    H[gid] = acc;
}

// ---------------- WMMA f32 GEMM: H(Mx128) = A(Mx128) @ W(128x128) ----------------
// Block = 256 threads = 8 waves; block covers 128 rows (16 per wave).
// W staged transposed in LDS so each lane's B fetch (K-pair, fixed N) is contiguous.
__global__ __launch_bounds__(256) void gemm128_wmma_kernel(const float* __restrict__ A,
                                                           const float* __restrict__ W,
                                                           float* __restrict__ H, int M) {
    __shared__ float Wt[HIDF * HIDF];   // Wt[n][k] = W[k][n], 64 KB
    for (int idx = threadIdx.x; idx < HIDF * HIDF; idx += 256) {
        int k = idx >> 7, n = idx & 127;
        Wt[n * HIDF + k] = W[idx];
    }
    __syncthreads();

    int wave = threadIdx.x >> 5;
    int lane = threadIdx.x & 31;
    int m    = lane & 15;          // row within A tile / col within B tile
    int half = lane >> 4;          // K-pair selector (A/B), M+8 selector (D)
    int row0 = (blockIdx.x * 8 + wave) * 16;
    if (row0 >= M) return;         // M is a multiple of 16: tiles are exact

    v8f c[8];
#pragma unroll
    for (int nt = 0; nt < 8; ++nt) c[nt] = (v8f){0,0,0,0,0,0,0,0};

    const float* arow = A + (size_t)(row0 + m) * HIDF;
    for (int k0 = 0; k0 < HIDF; k0 += 4) {
        int ka = k0 + 2 * half;
        v2f a = *(const v2f*)(arow + ka);                 // A[m][ka..ka+1]
#pragma unroll
        for (int nt = 0; nt < 8; ++nt) {
            int n = nt * 16 + m;
            v2f b = *(const v2f*)(&Wt[n * HIDF + ka]);    // W[ka..ka+1][n]
            c[nt] = __builtin_amdgcn_wmma_f32_16x16x4_f32(
                false, a, false, b, (short)0, c[nt], false, false);
        }
    }

    // D layout: VGPR r -> (M=r, N=lane) lanes 0-15 ; (M=r+8, N=lane-16) lanes 16-31
#pragma unroll
    for (int nt = 0; nt < 8; ++nt) {
        float* hrow = H + (size_t)(row0 + 8 * half) * HIDF + nt * 16 + m;
#pragma unroll
        for (int r = 0; r < 8; ++r) hrow[(size_t)r * HIDF] = c[nt][r];
    }
}

// ---------------- edge scatter: agg[dst] += H[src] * w ----------------
// One wave per edge, each lane handles 4 contiguous floats (16 B load, 4 f32 atomics).
__global__ __launch_bounds__(256) void scatter_kernel(const int* __restrict__ src,
                                                      const int* __restrict__ dst,
                                                      const float* __restrict__ w,
                                                      const float* __restrict__ H,
                                                      float* __restrict__ agg, int E) {
    int gid  = blockIdx.x * blockDim.x + threadIdx.x;
    int e    = gid >> 5;
    int lane = gid & 31;
    if (e >= E) return;
    int s = src[e], d = dst[e];
    float we = w[e];
    float4 v = *((const float4*)(H + (size_t)s * HIDF) + lane);
    float* op = agg + (size_t)d * HIDF + lane * 4;
    unsafeAtomicAdd(op + 0, v.x * we);
    unsafeAtomicAdd(op + 1, v.y * we);
    unsafeAtomicAdd(op + 2, v.z * we);
    unsafeAtomicAdd(op + 3, v.w * we);
}

// ---------------- finalize: agg = [relu](agg + H*invdeg + bias) ----------------
__global__ __launch_bounds__(256) void finalize_kernel(float* __restrict__ agg,
                                                       const float* __restrict__ H,
                                                       const float* __restrict__ invdeg,
                                                       const float* __restrict__ bias,
                                                       int M, int relu) {
    int gid = blockIdx.x * blockDim.x + threadIdx.x;
    if (gid >= M * HIDF) return;
    int i = gid >> 7, f = gid & 127;
    float v = agg[gid] + H[gid] * invdeg[i] + bias[f];
    agg[gid] = relu ? fmaxf(v, 0.0f) : v;
}

// ---------------- mean pool, stage 1: per-block LDS accumulation ----------------
__global__ __launch_bounds__(128) void pool_kernel(const float* __restrict__ h,
                                                   const int* __restrict__ batch,
                                                   float* __restrict__ pool,
                                                   float* __restrict__ cnt, int n) {
    __shared__ float acc[NG * HIDF];   // 32 KB
    __shared__ float ccnt[NG];
    int f = threadIdx.x;               // 0..127, owns feature column f
    for (int g = 0; g < NG; ++g) acc[g * HIDF + f] = 0.0f;
    if (f < NG) ccnt[f] = 0.0f;
    __syncthreads();
    int per = (n + gridDim.x - 1) / gridDim.x;
    int i0 = blockIdx.x * per;
    int i1 = min(n, i0 + per);
    for (int i = i0; i < i1; ++i) {
        int g = batch[i];
        acc[g * HIDF + f] += h[(size_t)i * HIDF + f];
        if (f == 0) ccnt[g] += 1.0f;
    }
    __syncthreads();
    for (int g = 0; g < NG; ++g) unsafeAtomicAdd(&pool[g * HIDF + f], acc[g * HIDF + f]);
    if (f < NG) unsafeAtomicAdd(&cnt[f], ccnt[f]);
}

// ---------------- final linear: out[g][o] = (pool[g]/cnt[g]) @ Wl + bl ----------------
__global__ __launch_bounds__(128) void linear_out_kernel(const float* __restrict__ pool,
                                                         const float* __restrict__ cnt,
                                                         const float* __restrict__ Wl,
                                                         const float* __restrict__ bl,
                                                         float* __restrict__ out) {
    int g = blockIdx.x;
    int o = threadIdx.x;
    if (o >= OUTF) return;
    float inv = 1.0f / fmaxf(cnt[g], 1.0f);
    float acc = bl[o];
    for (int f = 0; f < HIDF; ++f)
        acc = fmaf(pool[g * HIDF + f] * inv, Wl[f * OUTF + o], acc);
    out[g * OUTF + o] = acc;
}

// ---------------- driver ----------------
extern "C" void kernel_launch(void* const* d_in, const int* in_sizes, int n_in,
                              void* d_out, int out_size, void* d_ws, size_t ws_size,
                              hipStream_t stream) {
    const float* x     = (const float*)d_in[0];
    const int*   ei    = (const int*)d_in[1];      // (2, E) int32
    const int*   batch = (const int*)d_in[2];
    const float* W1 = (const float*)d_in[3];  const float* b1 = (const float*)d_in[4];
    const float* W2 = (const float*)d_in[5];  const float* b2 = (const float*)d_in[6];
    const float* W3 = (const float*)d_in[7];  const float* b3 = (const float*)d_in[8];
    const float* Wl = (const float*)d_in[9];  const float* bl = (const float*)d_in[10];
    const int* src = ei;
    const int* dst = ei + NE;

    // workspace carve (256 B aligned)
    char* p = (char*)d_ws;
    auto carve = [&p](size_t bytes) { void* r = (void*)p; p += (bytes + 255) & ~(size_t)255; return r; };
    float* buf0   = (float*)carve((size_t)NN * HIDF * 4);
    float* buf1   = (float*)carve((size_t)NN * HIDF * 4);
    float* buf2   = (float*)carve((size_t)NN * HIDF * 4);
    float* deg    = (float*)carve((size_t)NN * 4);
    float* dis    = (float*)carve((size_t)NN * 4);
    float* invdeg = (float*)carve((size_t)NN * 4);
    float* ew     = (float*)carve((size_t)NE * 4);
    float* pool   = (float*)carve((size_t)NG * HIDF * 4);
    float* cnt    = (float*)carve((size_t)NG * 4);

    const int nf  = NN * HIDF;                 // 12.8M
    const int gf  = (nf + 255) / 256;
    const int gE  = (NE + 255) / 256;
    const int gN  = (NN + 255) / 256;
    const int gSc = ((NE * 32) + 255) / 256;   // wave per edge
    const int gGm = ((NN / 16) + 7) / 8;       // 8 wave-tiles per block

    // gcn_norm (shared by all 3 layers)
    fill_kernel<<<gN, 256, 0, stream>>>(deg, 1.0f, NN);               // self-loop
    degree_kernel<<<gE, 256, 0, stream>>>(dst, deg, NE);
    norm_kernel<<<gN, 256, 0, stream>>>(deg, dis, invdeg, NN);
    edgew_kernel<<<gE, 256, 0, stream>>>(src, dst, dis, ew, NE);

    // layer 1: H1=x@W1 -> buf0 ; agg -> buf1 ; buf1 = relu(agg + H1*invdeg + b1)
    gemm_in_kernel<<<gf, 256, 0, stream>>>(x, W1, buf0, NN);
    fill_kernel<<<gf, 256, 0, stream>>>(buf1, 0.0f, nf);
    scatter_kernel<<<gSc, 256, 0, stream>>>(src, dst, ew, buf0, buf1, NE);
    finalize_kernel<<<gf, 256, 0, stream>>>(buf1, buf0, invdeg, b1, NN, 1);

    // layer 2: H2=buf1@W2 -> buf0 ; agg -> buf2 ; buf2 = relu(...)
    gemm128_wmma_kernel<<<gGm, 256, 0, stream>>>(buf1, W2, buf0, NN);
    fill_kernel<<<gf, 256, 0, stream>>>(buf2, 0.0f, nf);
    scatter_kernel<<<gSc, 256, 0, stream>>>(src, dst, ew, buf0, buf2, NE);
    finalize_kernel<<<gf, 256, 0, stream>>>(buf2, buf0, invdeg, b2, NN, 1);

    // layer 3: H3=buf2@W3 -> buf1 ; agg -> buf0 ; buf0 = (no relu)
    gemm128_wmma_kernel<<<gGm, 256, 0, stream>>>(buf2, W3, buf1, NN);
    fill_kernel<<<gf, 256, 0, stream>>>(buf0, 0.0f, nf);
    scatter_kernel<<<gSc, 256, 0, stream>>>(src, dst, ew, buf1, buf0, NE);
    finalize_kernel<<<gf, 256, 0, stream>>>(buf0, buf1, invdeg, b3, NN, 0);

    // mean pool + classifier
    fill_kernel<<<(NG * HIDF + NG + 255) / 256, 256, 0, stream>>>(pool, 0.0f, NG * HIDF + NG);
    pool_kernel<<<256, 128, 0, stream>>>(buf0, batch, pool, cnt, NN);
    linear_out_kernel<<<NG, 128, 0, stream>>>(pool, cnt, Wl, bl, (float*)d_out);
}